// RPNPredictor_65781719106269
// MI455X (gfx1250) — compile-verified
//
#include <hip/hip_runtime.h>
#include <stdint.h>
#include <math.h>

// ---------------- static problem config (must match reference) ----------------
#define BATCH    4
#define A_TOTAL  159882
#define K_TOTAL  4507          // 1000+1000+1000+1000+507
#define NSORT    8192          // pow2 >= K_TOTAL for bitonic sort
#define POST_NMS 1000
#define NMS_T    0.7f
#define MIN_SZ   0.001f
#define IMG_SZ   800.0f
#define BBOX_CLIP 4.1351665567f   // log(1000/16)
#define LVL_OFF_F 1000.0f

__constant__ int c_lvl_off[5] = {0, 120000, 150000, 157500, 159375};
__constant__ int c_lvl_n[5]   = {120000, 30000, 7500, 1875, 507};
__constant__ int c_sel_off[5] = {0, 1000, 2000, 3000, 4000};
__constant__ int c_sel_k[5]   = {1000, 1000, 1000, 1000, 507};

// float -> order-preserving unsigned key (descending sort = larger key first)
__device__ __forceinline__ unsigned f2key(float f) {
  unsigned u = __float_as_uint(f);
  return (u & 0x80000000u) ? ~u : (u | 0x80000000u);
}

// ============================================================================
// Tensor Data Mover: 1-D tile (nelem 4-byte words) Global -> LDS.
// D# layout per CDNA5 ISA ch.8: group0 {count=1, lds_addr, global_addr, type=2},
// group1 {data_size=4B, tensor_dim0=tile_dim0=nelem, tensor_dim1=1, stride}.
// Groups 2/3 unused (<=2D tensor). Tracked by TENSORcnt.
// ============================================================================
typedef __attribute__((ext_vector_type(4))) unsigned tdm_v4u;
typedef __attribute__((ext_vector_type(8))) int      tdm_v8i;
typedef __attribute__((ext_vector_type(4))) int      tdm_v4i;

__device__ __forceinline__ void tdm_load_1d(unsigned lds_off,
                                            unsigned long long gaddr,
                                            unsigned nelem) {
  tdm_v4u g0 = {
      1u,                                            // count=1, user descriptor
      lds_off,                                       // LDS byte address
      (unsigned)(gaddr & 0xFFFFFFFFull),             // global_addr[31:0]
      (unsigned)((gaddr >> 32) & 0x1FFFFFFull) | (2u << 30)  // [56:32] + type=2
  };
  tdm_v8i g1 = {
      (int)(2u << 16),                               // data_size=2 (4 bytes)
      (int)((nelem & 0xFFFFu) << 16),                // tensor_dim0[15:0] @bit48
      (int)(((nelem >> 16) & 0xFFFFu) | (1u << 16)), // tensor_dim0[31:16], dim1=1
      (int)((nelem & 0xFFFFu) << 16),                // tile_dim0 @bit112
      0,                                             // tile_dim1=0, tile_dim2=0
      (int)nelem,                                    // tensor_dim0_stride lo32
      0, 0
  };
  tdm_v4i z4 = {0, 0, 0, 0};
#if defined(__clang_major__) && (__clang_major__ >= 23)
  tdm_v8i z8 = {0, 0, 0, 0, 0, 0, 0, 0};
  __builtin_amdgcn_tensor_load_to_lds(g0, g1, z4, z4, z8, 0);
#else
  __builtin_amdgcn_tensor_load_to_lds(g0, g1, z4, z4, 0);
#endif
}

__device__ __forceinline__ unsigned lds_offset_of(const void* p) {
  return (unsigned)(unsigned long long)(uintptr_t)p;   // low 32 bits = LDS byte offset
}

// ============================================================================
// Kernel 1: exact per-(image,level) top-k via 4-pass radix select.
// One 1024-thread workgroup per (b, level). Order within the selected set is
// irrelevant: a global score sort follows.
// ============================================================================
__global__ void k_topk_select(const float* __restrict__ obj, int* __restrict__ sel_idx) {
  const int b = blockIdx.x / 5, lvl = blockIdx.x % 5;
  const int off = c_lvl_off[lvl], n = c_lvl_n[lvl], k = c_sel_k[lvl];
  const float* base = obj + (size_t)b * A_TOTAL + off;
  const int nt = (int)blockDim.x;

  __shared__ unsigned hist[256];
  __shared__ unsigned s_prefix, s_remaining;
  if (threadIdx.x == 0) { s_prefix = 0u; s_remaining = (unsigned)k; }
  __syncthreads();

  for (int pass = 0; pass < 4; ++pass) {
    const int shift = 24 - pass * 8;
    const unsigned prefix = s_prefix;
    const unsigned himask = (pass == 0) ? 0u : (0xFFFFFFFFu << (shift + 8));
    if (threadIdx.x < 256) hist[threadIdx.x] = 0u;
    __syncthreads();
    for (int i = (int)threadIdx.x; i < n; i += nt) {
      __builtin_prefetch(base + i + 8192, 0, 0);   // global_prefetch_b8 (speculative)
      unsigned key = f2key(base[i]);
      if ((key & himask) == prefix)
        atomicAdd(&hist[(key >> shift) & 0xFFu], 1u);
    }
    __syncthreads();
    if (threadIdx.x == 0) {
      unsigned rem = s_remaining, cum = 0u;
      int bin = 255;
      for (; bin > 0; --bin) {
        unsigned c = hist[bin];
        if (cum + c >= rem) break;
        cum += c;
      }
      s_prefix    = prefix | ((unsigned)bin << shift);
      s_remaining = rem - cum;
    }
    __syncthreads();
  }

  const unsigned T   = s_prefix;          // exact k-th largest key
  const unsigned rem = s_remaining;       // how many == T to take
  const int num_gt   = k - (int)rem;
  __shared__ unsigned cnt_gt, cnt_eq;
  if (threadIdx.x == 0) { cnt_gt = 0u; cnt_eq = 0u; }
  __syncthreads();

  int* out = sel_idx + b * K_TOTAL + c_sel_off[lvl];
  for (int i = (int)threadIdx.x; i < n; i += nt) {
    unsigned key = f2key(base[i]);
    if (key > T) {
      unsigned p = atomicAdd(&cnt_gt, 1u);
      out[p] = off + i;
    } else if (key == T) {
      unsigned p = atomicAdd(&cnt_eq, 1u);
      if (p < rem) out[num_gt + (int)p] = off + i;
    }
  }
}

// ============================================================================
// Kernel 2: gather selected anchors, sigmoid, box decode, clip, validity,
// emit sort keys (pad slots [K_TOTAL,NSORT) get key 0).
// ============================================================================
__global__ void k_decode(const float* __restrict__ obj,
                         const float* __restrict__ deltas,
                         const float* __restrict__ anchors,
                         const int* __restrict__ sel_idx,
                         float* __restrict__ boxes_raw,   // [B][K_TOTAL][4]
                         float* __restrict__ score_arr,   // [B][K_TOTAL]
                         int*   __restrict__ lvl_arr,     // [B][K_TOTAL]
                         unsigned* __restrict__ keys) {   // [B][NSORT]
  const int t = blockIdx.x * blockDim.x + threadIdx.x;
  const int b = t / NSORT, s = t % NSORT;
  if (b >= BATCH) return;
  if (s >= K_TOTAL) { keys[b * NSORT + s] = 0u; return; }

  const int lvl = (s >= 4000) ? 4 : (s / 1000);
  const int i   = sel_idx[b * K_TOTAL + s];

  const float logit = obj[(size_t)b * A_TOTAL + i];
  const float sc = 1.0f / (1.0f + __expf(-logit));

  const float4 a = ((const float4*)anchors)[i];                       // b128
  const float wa = a.z - a.x, ha = a.w - a.y;
  const float cxa = a.x + 0.5f * wa, cya = a.y + 0.5f * ha;

  const float4 d = ((const float4*)deltas)[(size_t)b * A_TOTAL + i];  // b128
  const float dw = fminf(d.z, BBOX_CLIP), dh = fminf(d.w, BBOX_CLIP);
  const float cx = d.x * wa + cxa, cy = d.y * ha + cya;
  const float w  = __expf(dw) * wa, h = __expf(dh) * ha;

  float x1 = fminf(fmaxf(cx - 0.5f * w, 0.0f), IMG_SZ);
  float y1 = fminf(fmaxf(cy - 0.5f * h, 0.0f), IMG_SZ);
  float x2 = fminf(fmaxf(cx + 0.5f * w, 0.0f), IMG_SZ);
  float y2 = fminf(fmaxf(cy + 0.5f * h, 0.0f), IMG_SZ);

  const bool valid = ((x2 - x1) >= MIN_SZ) && ((y2 - y1) >= MIN_SZ); // score>=0 always (sigmoid)

  ((float4*)boxes_raw)[(size_t)b * K_TOTAL + s] = make_float4(x1, y1, x2, y2);
  score_arr[b * K_TOTAL + s] = sc;
  lvl_arr[b * K_TOTAL + s]   = lvl;
  keys[b * NSORT + s] = valid ? f2key(sc) : 0u;   // invalid -> smallest key
}

// ============================================================================
// Kernel 3: per-image bitonic sort (descending) of NSORT (key,payload) pairs
// in LDS (keys staged by TDM), then write sorted raw boxes / offset boxes /
// scores / valid flags.
// ============================================================================
__global__ void k_sort(const unsigned* __restrict__ keys,
                       const float* __restrict__ boxes_raw,
                       const float* __restrict__ score_arr,
                       const int*   __restrict__ lvl_arr,
                       float* __restrict__ sboxes_off,  // [B][K_TOTAL][4] (+lvl*1000)
                       float* __restrict__ sraw,        // [B][K_TOTAL][4]
                       float* __restrict__ sscore,      // [B][K_TOTAL]
                       int*   __restrict__ svalid) {    // [B][K_TOTAL]
  const int b = blockIdx.x;
  __shared__ unsigned sk[NSORT];
  __shared__ unsigned sp[NSORT];

  // TDM: DMA the 32KB key tile into LDS; init payloads as overlap work.
  if (threadIdx.x == 0) {
    tdm_load_1d(lds_offset_of(sk),
                (unsigned long long)(uintptr_t)(keys + (size_t)b * NSORT),
                (unsigned)NSORT);
  }
  for (int i = (int)threadIdx.x; i < NSORT; i += (int)blockDim.x) sp[i] = (unsigned)i;
  if (threadIdx.x == 0) __builtin_amdgcn_s_wait_tensorcnt(0);
  __syncthreads();

  for (int k = 2; k <= NSORT; k <<= 1) {
    for (int j = k >> 1; j > 0; j >>= 1) {
      for (int i = (int)threadIdx.x; i < NSORT; i += (int)blockDim.x) {
        const int p = i ^ j;
        if (p > i) {
          const bool desc = ((i & k) == 0);
          const unsigned ki = sk[i], kp = sk[p];
          const bool swap = desc ? (ki < kp) : (ki > kp);
          if (swap) {
            sk[i] = kp; sk[p] = ki;
            const unsigned tmp = sp[i]; sp[i] = sp[p]; sp[p] = tmp;
          }
        }
      }
      __syncthreads();
    }
  }

  for (int t = (int)threadIdx.x; t < K_TOTAL; t += (int)blockDim.x) {
    const unsigned src = sp[t];
    const bool ok = (sk[t] != 0u) && (src < (unsigned)K_TOTAL);
    float4* so = (float4*)sboxes_off + (size_t)b * K_TOTAL + t;
    float4* sr = (float4*)sraw       + (size_t)b * K_TOTAL + t;
    if (ok) {
      const float4 rb = ((const float4*)boxes_raw)[(size_t)b * K_TOTAL + src];
      const float offv = (float)lvl_arr[b * K_TOTAL + src] * LVL_OFF_F;
      *sr = rb;
      *so = make_float4(rb.x + offv, rb.y + offv, rb.z + offv, rb.w + offv);
      sscore[b * K_TOTAL + t] = score_arr[b * K_TOTAL + src];
    } else {
      *sr = make_float4(0.f, 0.f, 0.f, 0.f);
      *so = make_float4(0.f, 0.f, 0.f, 0.f);
      sscore[b * K_TOTAL + t] = 0.0f;
    }
    svalid[b * K_TOTAL + t] = ok ? 1 : 0;
  }
}

// ============================================================================
// Kernel 4: greedy batched NMS per image + rank scan + output scatter.
// One 1024-thread workgroup per image. Offset boxes staged into LDS by TDM
// (issued first, waited after the overlap work). Boxes read as float4
// (ds_load_b128); per-box areas precomputed in LDS.
// Dynamic LDS: K*16 (boxes) + K*4 (area) + K*4 (keep) + 4KB (scan) = ~112KB.
// ============================================================================
__global__ void k_nms(const float* __restrict__ sboxes_off,
                      const float* __restrict__ sraw,
                      const float* __restrict__ sscore,
                      const int*   __restrict__ svalid,
                      float* __restrict__ out) {
  const int b = blockIdx.x;
  extern __shared__ float smem[];
  float4*   bx4  = (float4*)smem;                         // [K_TOTAL]
  float*    area = smem + (size_t)K_TOTAL * 4;            // [K_TOTAL]
  int*      keep = (int*)(area + K_TOTAL);                // [K_TOTAL]
  unsigned* ps   = (unsigned*)(keep + K_TOTAL);           // [1024]

  // TDM: DMA the 72KB offset-box tile into LDS first (overlaps work below)
  if (threadIdx.x == 0) {
    tdm_load_1d(lds_offset_of(bx4),
                (unsigned long long)(uintptr_t)(sboxes_off + (size_t)b * K_TOTAL * 4),
                (unsigned)(K_TOTAL * 4));
  }

  // overlap: zero the output slice + load keep flags
  float* outb = out + (size_t)b * POST_NMS * 5;
  for (int i = (int)threadIdx.x; i < POST_NMS * 5; i += (int)blockDim.x) outb[i] = 0.0f;
  for (int i = (int)threadIdx.x; i < K_TOTAL; i += (int)blockDim.x)
    keep[i] = svalid[b * K_TOTAL + i];

  if (threadIdx.x == 0) __builtin_amdgcn_s_wait_tensorcnt(0);
  __syncthreads();

  // precompute areas (boxes now resident in LDS)
  for (int i = (int)threadIdx.x; i < K_TOTAL; i += (int)blockDim.x) {
    const float4 bb = bx4[i];
    area[i] = (bb.z - bb.x) * (bb.w - bb.y);
  }
  __syncthreads();

  // greedy suppression over score-sorted entries
  for (int i = 0; i < K_TOTAL; ++i) {
    __syncthreads();
    if (keep[i]) {                       // uniform (same LDS word for all threads)
      const float4 bi = bx4[i];
      const float  ai = area[i];
      for (int j = i + 1 + (int)threadIdx.x; j < K_TOTAL; j += (int)blockDim.x) {
        if (keep[j]) {
          const float4 bj = bx4[j];                      // ds_load_b128
          const float xx1 = fmaxf(bi.x, bj.x);
          const float yy1 = fmaxf(bi.y, bj.y);
          const float xx2 = fminf(bi.z, bj.z);
          const float yy2 = fminf(bi.w, bj.w);
          const float w = fmaxf(xx2 - xx1, 0.0f);
          const float h = fmaxf(yy2 - yy1, 0.0f);
          const float inter = w * h;
          const float iou = inter / (ai + area[j] - inter);
          if (iou > NMS_T) keep[j] = 0;
        }
      }
    }
  }
  __syncthreads();

  // rank survivors: blocked prefix scan (chunk of 5 per thread, 1024 partials)
  const int CH = 5;                          // ceil(4507/1024)
  const int s0 = (int)threadIdx.x * CH;
  const int s1 = (s0 + CH < K_TOTAL) ? (s0 + CH) : K_TOTAL;
  unsigned loc = 0u;
  for (int e = s0; e < s1; ++e) loc += (keep[e] ? 1u : 0u);
  ps[threadIdx.x] = loc;
  __syncthreads();
  for (int d = 1; d < 1024; d <<= 1) {
    unsigned v = (threadIdx.x >= (unsigned)d) ? ps[threadIdx.x - d] : 0u;
    __syncthreads();
    ps[threadIdx.x] += v;
    __syncthreads();
  }
  unsigned rank = ps[threadIdx.x] - loc;     // exclusive base for this chunk

  for (int e = s0; e < s1; ++e) {
    if (keep[e]) {
      const unsigned r = rank++;
      if (r < POST_NMS) {
        const float4 rb = ((const float4*)sraw)[(size_t)b * K_TOTAL + e];
        float* o = outb + (size_t)r * 5;
        o[0] = rb.x; o[1] = rb.y; o[2] = rb.z; o[3] = rb.w;
        o[4] = sscore[b * K_TOTAL + e];
      }
    }
  }
}

// ============================================================================
extern "C" void kernel_launch(void* const* d_in, const int* in_sizes, int n_in,
                              void* d_out, int out_size, void* d_ws, size_t ws_size,
                              hipStream_t stream) {
  (void)in_sizes; (void)n_in; (void)out_size; (void)ws_size;
  const float* obj     = (const float*)d_in[0];  // [B, A_TOTAL]
  const float* deltas  = (const float*)d_in[1];  // [B, A_TOTAL, 4]
  const float* anchors = (const float*)d_in[2];  // [A_TOTAL, 4]
  float* out = (float*)d_out;                    // [B, POST_NMS, 5]

  char* ws = (char*)d_ws;
  size_t o = 0;
  auto take = [&](size_t bytes) -> void* {
    o = (o + 255) & ~(size_t)255;
    void* p = ws + o;
    o += bytes;
    return p;
  };
  int*      sel_idx    = (int*)     take((size_t)BATCH * K_TOTAL * sizeof(int));
  unsigned* keys       = (unsigned*)take((size_t)BATCH * NSORT   * sizeof(unsigned));
  float*    boxes_raw  = (float*)   take((size_t)BATCH * K_TOTAL * 4 * sizeof(float));
  float*    score_arr  = (float*)   take((size_t)BATCH * K_TOTAL * sizeof(float));
  int*      lvl_arr    = (int*)     take((size_t)BATCH * K_TOTAL * sizeof(int));
  float*    sboxes_off = (float*)   take((size_t)BATCH * K_TOTAL * 4 * sizeof(float));
  float*    sraw       = (float*)   take((size_t)BATCH * K_TOTAL * 4 * sizeof(float));
  float*    sscore     = (float*)   take((size_t)BATCH * K_TOTAL * sizeof(float));
  int*      svalid     = (int*)     take((size_t)BATCH * K_TOTAL * sizeof(int));

  k_topk_select<<<BATCH * 5, 1024, 0, stream>>>(obj, sel_idx);
  k_decode<<<(BATCH * NSORT + 255) / 256, 256, 0, stream>>>(
      obj, deltas, anchors, sel_idx, boxes_raw, score_arr, lvl_arr, keys);
  k_sort<<<BATCH, 1024, 0, stream>>>(
      keys, boxes_raw, score_arr, lvl_arr, sboxes_off, sraw, sscore, svalid);
  const size_t shbytes = (size_t)K_TOTAL * 4 * sizeof(float)   // boxes
                       + (size_t)K_TOTAL * sizeof(float)       // area
                       + (size_t)K_TOTAL * sizeof(int)         // keep
                       + 1024 * sizeof(unsigned);              // scan
  k_nms<<<BATCH, 1024, shbytes, stream>>>(sboxes_off, sraw, sscore, svalid, out);
}